// Generator_pts_13108240187429
// MI455X (gfx1250) — compile-verified
//
#include <hip/hip_runtime.h>
#include <hip/hip_bf16.h>

typedef __attribute__((ext_vector_type(2))) float v2f;
typedef __attribute__((ext_vector_type(8))) float v8f;

#define BN_SCALEF 0.9999950000374997f  // 1/sqrt(1+1e-5)
#define FPS_T 1024

// ---------------------------------------------------------------------------
// Farthest point sampling. One block per batch item. Each thread caches its
// strided subset of points (<=16) plus running min-distance in registers.
// Argmax reduction: wave32 shuffle tree, then one LDS pass over 32 partials.
// ---------------------------------------------------------------------------
__global__ void fps_kernel(const float* __restrict__ xyz,
                           float* __restrict__ new_xyz,
                           int N, int npoint) {
  const int b = blockIdx.x;
  const int tid = threadIdx.x;
  const int lane = tid & 31;
  const int wid = tid >> 5;
  const float* P = xyz + (size_t)b * N * 3;

  float px[16], py[16], pz[16], pd[16];
  int cnt = 0;
  for (int j = tid; j < N; j += FPS_T) {
    px[cnt] = P[j * 3 + 0];
    py[cnt] = P[j * 3 + 1];
    pz[cnt] = P[j * 3 + 2];
    pd[cnt] = 1e10f;
    ++cnt;
  }

  __shared__ float swav[32];
  __shared__ int swai[32];
  __shared__ float scen[3];
  if (tid == 0) {
    scen[0] = P[0]; scen[1] = P[1]; scen[2] = P[2];
  }
  __syncthreads();

  for (int k = 0; k < npoint; ++k) {
    float cx = scen[0], cy = scen[1], cz = scen[2];
    if (tid == 0) {
      new_xyz[((size_t)b * npoint + k) * 3 + 0] = cx;
      new_xyz[((size_t)b * npoint + k) * 3 + 1] = cy;
      new_xyz[((size_t)b * npoint + k) * 3 + 2] = cz;
    }
    float bv = -1.0f;
    int bi = N;
    for (int q = 0; q < cnt; ++q) {
      float dx = px[q] - cx, dy = py[q] - cy, dz = pz[q] - cz;
      float d = dx * dx + dy * dy + dz * dz;
      float nd = fminf(pd[q], d);
      pd[q] = nd;
      if (nd > bv) { bv = nd; bi = tid + q * FPS_T; }
    }
    // wave32 shuffle argmax (tie -> smaller index, matching jnp.argmax)
    for (int o = 16; o > 0; o >>= 1) {
      float ov = __shfl_down(bv, o, 32);
      int oi = __shfl_down(bi, o, 32);
      if (ov > bv || (ov == bv && oi < bi)) { bv = ov; bi = oi; }
    }
    if (lane == 0) { swav[wid] = bv; swai[wid] = bi; }
    __syncthreads();
    if (tid < 32) {
      bv = swav[tid];
      bi = swai[tid];
      for (int o = 16; o > 0; o >>= 1) {
        float ov = __shfl_down(bv, o, 32);
        int oi = __shfl_down(bi, o, 32);
        if (ov > bv || (ov == bv && oi < bi)) { bv = ov; bi = oi; }
      }
      if (tid == 0) {
        scen[0] = P[bi * 3 + 0];
        scen[1] = P[bi * 3 + 1];
        scen[2] = P[bi * 3 + 2];
      }
    }
    __syncthreads();
  }
}

// ---------------------------------------------------------------------------
// Ball query + grouping, fused. One wave (32 lanes) per query point.
// Ordered compaction via wave32 ballot/popc reproduces the reference's
// "first nsample indices (ascending) within radius" semantics, including
// pad-with-first and the empty->N-1 case. Then gathers relative xyz + feats
// into g[b][C][np][ns].
// ---------------------------------------------------------------------------
__global__ void ballquery_group_kernel(const float* __restrict__ xyz, int Nsrc,
                                       const float* __restrict__ nxyz, int np,
                                       const float* __restrict__ feats, int Cf,
                                       float* __restrict__ g, int ns, float r2,
                                       int C /* = 3 + Cf */) {
  const int b = blockIdx.y;
  const int i = blockIdx.x;
  const int lane = threadIdx.x;
  const float* P = xyz + (size_t)b * Nsrc * 3;
  const float cx = nxyz[((size_t)b * np + i) * 3 + 0];
  const float cy = nxyz[((size_t)b * np + i) * 3 + 1];
  const float cz = nxyz[((size_t)b * np + i) * 3 + 2];

  __shared__ int sel[32];
  int cnt = 0;
  for (int j0 = 0; j0 < Nsrc && cnt < ns; j0 += 32) {
    int j = j0 + lane;
    bool hit = false;
    if (j < Nsrc) {
      float dx = P[j * 3 + 0] - cx;
      float dy = P[j * 3 + 1] - cy;
      float dz = P[j * 3 + 2] - cz;
      hit = (dx * dx + dy * dy + dz * dz) < r2;
    }
    unsigned mask = (unsigned)__ballot(hit);
    int pos = cnt + __popc(mask & ((1u << lane) - 1u));
    if (hit && pos < ns) sel[pos] = j;
    cnt += __popc(mask);
  }
  if (cnt > ns) cnt = ns;
  __syncthreads();
  if (lane < ns && lane >= cnt) {
    sel[lane] = (cnt > 0) ? sel[0] : (Nsrc - 1);
  }
  __syncthreads();

  if (lane < ns) {
    int j = sel[lane];
    float gx = P[j * 3 + 0] - cx;
    float gy = P[j * 3 + 1] - cy;
    float gz = P[j * 3 + 2] - cz;
    g[(((size_t)b * C + 0) * np + i) * ns + lane] = gx;
    g[(((size_t)b * C + 1) * np + i) * ns + lane] = gy;
    g[(((size_t)b * C + 2) * np + i) * ns + lane] = gz;
    for (int c = 0; c < Cf; ++c) {
      g[(((size_t)b * C + 3 + c) * np + i) * ns + lane] =
          feats[((size_t)b * Cf + c) * Nsrc + j];
    }
  }
}

// ---------------------------------------------------------------------------
// Pointwise-conv GEMM: Y[b,m,col] = act((W X + b0) * (scale*BN_SCALE) + b1)
// One wave owns a 16x64 output tile: four V_WMMA_F32_16X16X4_F32 chains
// sharing one A (weight) fragment per k-step. Requires NCOL % 64 == 0
// (true for every layer in this network). Hot loop is branchless: row guard
// = clamped address + cndmask-zero; K%4 remainder is a clamped tail step.
// A layout: lane(0-15)=row M, vgprs hold K,K+1; lanes 16-31 hold K+2,K+3.
// B layout mirrors with N across lanes. D: vgpr r -> rows r / r+8.
// ---------------------------------------------------------------------------
__global__ void mlp_wmma_kernel(const float* __restrict__ X,
                                const float* __restrict__ W,
                                const float* __restrict__ b0,
                                const float* __restrict__ scale,
                                const float* __restrict__ b1,
                                float* __restrict__ Y,
                                int M, int K, int NCOL, int relu) {
  const int b = blockIdx.y;
  const float* Xb = X + (size_t)b * K * NCOL;
  float* Yb = Y + (size_t)b * M * NCOL;

  const int wave = threadIdx.x >> 5;
  const int lane = threadIdx.x & 31;
  const int tilesN = NCOL >> 6;                // 16x64 tiles
  const int tilesM = (M + 15) >> 4;
  const int tile = blockIdx.x * (blockDim.x >> 5) + wave;
  if (tile >= tilesM * tilesN) return;
  const int tm = tile / tilesN;
  const int tn = tile - tm * tilesN;
  const int m0 = tm << 4;
  const int n0 = tn << 6;

  const int lr = lane & 15;
  const int half = lane >> 4;
  const int koff = half << 1;                  // 0 or 2
  const int mrow = m0 + lr;
  const bool mok = (mrow < M);
  const int arow = mok ? mrow : 0;             // clamped row (always valid)
  const int col0 = n0 + lr;

  const int K4 = K & ~3;
  const float* wq = W + (size_t)arow * K + koff;
  const float* xq = Xb + (size_t)koff * NCOL + col0;
  const size_t xstep = (size_t)4 * NCOL;

  v8f acc0 = {0.f, 0.f, 0.f, 0.f, 0.f, 0.f, 0.f, 0.f};
  v8f acc1 = {0.f, 0.f, 0.f, 0.f, 0.f, 0.f, 0.f, 0.f};
  v8f acc2 = {0.f, 0.f, 0.f, 0.f, 0.f, 0.f, 0.f, 0.f};
  v8f acc3 = {0.f, 0.f, 0.f, 0.f, 0.f, 0.f, 0.f, 0.f};

#pragma unroll 2
  for (int k = 0; k < K4; k += 4) {
    float ax = wq[0];
    float ay = wq[1];
    if (!mok) { ax = 0.f; ay = 0.f; }          // v_cndmask, no branch
    v2f A = {ax, ay};
    v2f B0 = {xq[0], xq[NCOL]};
    v2f B1 = {xq[16], xq[NCOL + 16]};
    v2f B2 = {xq[32], xq[NCOL + 32]};
    v2f B3 = {xq[48], xq[NCOL + 48]};
    acc0 = __builtin_amdgcn_wmma_f32_16x16x4_f32(false, A, false, B0,
                                                 (short)0, acc0, false, false);
    acc1 = __builtin_amdgcn_wmma_f32_16x16x4_f32(false, A, false, B1,
                                                 (short)0, acc1, false, false);
    acc2 = __builtin_amdgcn_wmma_f32_16x16x4_f32(false, A, false, B2,
                                                 (short)0, acc2, false, false);
    acc3 = __builtin_amdgcn_wmma_f32_16x16x4_f32(false, A, false, B3,
                                                 (short)0, acc3, false, false);
    wq += 4;
    xq += xstep;
  }

  if (K4 < K) {  // K % 4 == 3 in this network; clamped branchless tail
    const int ka = K4 + koff;
    const int ka0 = (ka < K) ? ka : (K - 1);
    const int ka1 = (ka + 1 < K) ? (ka + 1) : (K - 1);
    const bool v0 = mok && (ka < K);
    const bool v1 = mok && (ka + 1 < K);
    const bool c0 = (ka < K);
    const bool c1 = (ka + 1 < K);
    float ax = W[(size_t)arow * K + ka0];
    float ay = W[(size_t)arow * K + ka1];
    if (!v0) ax = 0.f;
    if (!v1) ay = 0.f;
    v2f A = {ax, ay};
    const float* x0 = Xb + (size_t)ka0 * NCOL + col0;
    const float* x1 = Xb + (size_t)ka1 * NCOL + col0;
#pragma unroll
    for (int t = 0; t < 4; ++t) {
      float bx = x0[t * 16];
      float by = x1[t * 16];
      if (!c0) bx = 0.f;
      if (!c1) by = 0.f;
      v2f Bt = {bx, by};
      v8f* accp = (t == 0) ? &acc0 : (t == 1) ? &acc1 : (t == 2) ? &acc2 : &acc3;
      *accp = __builtin_amdgcn_wmma_f32_16x16x4_f32(false, A, false, Bt,
                                                    (short)0, *accp, false, false);
    }
  }

#pragma unroll
  for (int r = 0; r < 8; ++r) {
    const int m = m0 + r + (half << 3);
    if (m < M) {
      float s = scale ? (scale[m] * BN_SCALEF) : 1.f;
      float add0 = b0 ? b0[m] : 0.f;
      float add1 = b1 ? b1[m] : 0.f;
      float t0 = (acc0[r] + add0) * s + add1;
      float t1 = (acc1[r] + add0) * s + add1;
      float t2 = (acc2[r] + add0) * s + add1;
      float t3 = (acc3[r] + add0) * s + add1;
      if (relu) {
        t0 = fmaxf(t0, 0.f); t1 = fmaxf(t1, 0.f);
        t2 = fmaxf(t2, 0.f); t3 = fmaxf(t3, 0.f);
      }
      float* yrow = Yb + (size_t)m * NCOL + n0 + lr;
      yrow[0] = t0;
      yrow[16] = t1;
      yrow[32] = t2;
      yrow[48] = t3;
    }
  }
}

// ---------------------------------------------------------------------------
// Max over the sample dimension; writes into the concat feature buffer at
// channel offset 'coff' (layout [B][Ctot][np]).
// ---------------------------------------------------------------------------
__global__ void maxpool_kernel(const float* __restrict__ X,
                               float* __restrict__ Y,
                               int M, int np, int ns, int Ctot, int coff) {
  const int b = blockIdx.y;
  const int idx = blockIdx.x * blockDim.x + threadIdx.x;
  if (idx >= M * np) return;
  const int m = idx / np;
  const int i = idx - m * np;
  const float* src = X + (((size_t)b * M + m) * np + i) * ns;
  float v = src[0];
  for (int t = 1; t < ns; ++t) v = fmaxf(v, src[t]);
  Y[((size_t)b * Ctot + coff + m) * np + i] = v;
}

// ---------------------------------------------------------------------------
// 3-NN inverse-distance interpolation. One wave per unknown point: per-lane
// top-3 over strided known points, LDS merge, then lanes stride channels.
// Writes into concat buffer [B][Ccat][Nu] at channel offset 'coff'.
// ---------------------------------------------------------------------------
__global__ void three_interp_kernel(const float* __restrict__ uxyz, int Nu,
                                    const float* __restrict__ kxyz, int Nk,
                                    const float* __restrict__ feats, int C,
                                    float* __restrict__ dst, int Ccat, int coff) {
  const int b = blockIdx.y;
  const int i = blockIdx.x;
  const int lane = threadIdx.x;
  const float ux = uxyz[((size_t)b * Nu + i) * 3 + 0];
  const float uy = uxyz[((size_t)b * Nu + i) * 3 + 1];
  const float uz = uxyz[((size_t)b * Nu + i) * 3 + 2];
  const float* KP = kxyz + (size_t)b * Nk * 3;

  float d0 = 1e30f, d1 = 1e30f, d2 = 1e30f;
  int i0 = 0, i1 = 0, i2 = 0;
  for (int j = lane; j < Nk; j += 32) {
    float dx = KP[j * 3 + 0] - ux;
    float dy = KP[j * 3 + 1] - uy;
    float dz = KP[j * 3 + 2] - uz;
    float d = dx * dx + dy * dy + dz * dz;
    if (d < d0) { d2 = d1; i2 = i1; d1 = d0; i1 = i0; d0 = d; i0 = j; }
    else if (d < d1) { d2 = d1; i2 = i1; d1 = d; i1 = j; }
    else if (d < d2) { d2 = d; i2 = j; }
  }

  __shared__ float sd[96];
  __shared__ int si[96];
  __shared__ float sw[3];
  __shared__ int sj[3];
  sd[lane] = d0; sd[32 + lane] = d1; sd[64 + lane] = d2;
  si[lane] = i0; si[32 + lane] = i1; si[64 + lane] = i2;
  __syncthreads();
  if (lane == 0) {
    float g0 = 1e30f, g1 = 1e30f, g2 = 1e30f;
    int j0 = 0, j1 = 0, j2 = 0;
    for (int t = 0; t < 96; ++t) {
      float d = sd[t]; int j = si[t];
      if (d < g0) { g2 = g1; j2 = j1; g1 = g0; j1 = j0; g0 = d; j0 = j; }
      else if (d < g1) { g2 = g1; j2 = j1; g1 = d; j1 = j; }
      else if (d < g2) { g2 = d; j2 = j; }
    }
    float w0 = 1.f / (sqrtf(fmaxf(g0, 0.f)) + 1e-8f);
    float w1 = 1.f / (sqrtf(fmaxf(g1, 0.f)) + 1e-8f);
    float w2 = 1.f / (sqrtf(fmaxf(g2, 0.f)) + 1e-8f);
    float wsum = w0 + w1 + w2;
    sw[0] = w0 / wsum; sw[1] = w1 / wsum; sw[2] = w2 / wsum;
    sj[0] = j0; sj[1] = j1; sj[2] = j2;
  }
  __syncthreads();
  const float w0 = sw[0], w1 = sw[1], w2 = sw[2];
  const int j0 = sj[0], j1 = sj[1], j2 = sj[2];
  const float* F = feats + (size_t)b * C * Nk;
  for (int c = lane; c < C; c += 32) {
    float v = w0 * F[(size_t)c * Nk + j0] + w1 * F[(size_t)c * Nk + j1] +
              w2 * F[(size_t)c * Nk + j2];
    dst[((size_t)b * Ccat + coff + c) * Nu + i] = v;
  }
}

// Copy skip features into the low channels of the concat buffer.
__global__ void copychan_kernel(const float* __restrict__ src,
                                float* __restrict__ dst,
                                int C, int npts, int Ccat) {
  const int b = blockIdx.y;
  const int idx = blockIdx.x * blockDim.x + threadIdx.x;
  if (idx >= C * npts) return;
  const int c = idx / npts;
  const int i = idx - c * npts;
  dst[((size_t)b * Ccat + c) * npts + i] = src[((size_t)b * C + c) * npts + i];
}

// (B,3,N) -> (B,N,3)
__global__ void transpose_out_kernel(const float* __restrict__ src,
                                     float* __restrict__ dst, int N) {
  const int b = blockIdx.y;
  const int n = blockIdx.x * blockDim.x + threadIdx.x;
  if (n >= N) return;
  for (int o = 0; o < 3; ++o)
    dst[((size_t)b * N + n) * 3 + o] = src[((size_t)b * 3 + o) * N + n];
}

// ---------------------------------------------------------------------------
// Host orchestration
// ---------------------------------------------------------------------------
static const int kNs[5] = {16384, 4096, 1024, 256, 64};
static const int kCf[5] = {0, 96, 256, 512, 1024};  // feature chans per level
static const float kRadius[4][2] = {{0.1f, 0.5f}, {0.5f, 1.0f}, {1.0f, 2.0f}, {2.0f, 4.0f}};
static const int kNSample[2] = {16, 32};
static const int SA_CH[4][2][4] = {
    {{3, 16, 16, 32}, {3, 32, 32, 64}},
    {{99, 64, 64, 128}, {99, 64, 96, 128}},
    {{259, 128, 196, 256}, {259, 128, 196, 256}},
    {{515, 256, 256, 512}, {515, 256, 384, 512}}};
static const int FP_CH[4][3] = {
    {256, 128, 128}, {608, 256, 256}, {768, 512, 512}, {1536, 512, 512}};

extern "C" void kernel_launch(void* const* d_in, const int* in_sizes, int n_in,
                              void* d_out, int out_size, void* d_ws, size_t ws_size,
                              hipStream_t stream) {
  const int B = 2;
  (void)in_sizes; (void)n_in; (void)out_size; (void)ws_size;

  // ---- parse inputs (insertion order of setup_inputs dict) ----
  int p = 0;
  const float* PC = (const float*)d_in[p++];
  const float *saW[4][2][3], *saG[4][2][3], *saB[4][2][3];
  for (int l = 0; l < 4; ++l)
    for (int s = 0; s < 2; ++s)
      for (int li = 0; li < 3; ++li) {
        saW[l][s][li] = (const float*)d_in[p++];
        saG[l][s][li] = (const float*)d_in[p++];
        saB[l][s][li] = (const float*)d_in[p++];
      }
  const float *fpW[4][2], *fpG[4][2], *fpB[4][2];
  for (int k = 0; k < 4; ++k)
    for (int li = 0; li < 2; ++li) {
      fpW[k][li] = (const float*)d_in[p++];
      fpG[k][li] = (const float*)d_in[p++];
      fpB[k][li] = (const float*)d_in[p++];
    }
  const float* hW1 = (const float*)d_in[p++];
  const float* hb1 = (const float*)d_in[p++];
  const float* hg1 = (const float*)d_in[p++];
  const float* hbe1 = (const float*)d_in[p++];
  const float* hW2 = (const float*)d_in[p++];
  const float* hb2 = (const float*)d_in[p++];

  // ---- workspace layout (floats) ----
  float* ws = (float*)d_ws;
  size_t off = 0;
  auto alloc = [&](size_t n) { float* q = ws + off; off += n; return q; };

  float* nxw[5] = {nullptr, nullptr, nullptr, nullptr, nullptr};
  for (int l = 1; l <= 4; ++l) nxw[l] = alloc((size_t)B * kNs[l] * 3);
  float* lf[5] = {nullptr, nullptr, nullptr, nullptr, nullptr};
  for (int l = 1; l <= 4; ++l) lf[l] = alloc((size_t)B * kCf[l] * kNs[l]);
  const size_t BUF = (size_t)B * 64 * 4096 * 32;  // 16.78M floats
  float* bufA = alloc(BUF);
  float* bufB = alloc(BUF);
  float* fpOut[4];
  fpOut[3] = alloc((size_t)B * 512 * kNs[3]);
  fpOut[2] = alloc((size_t)B * 512 * kNs[2]);
  fpOut[1] = alloc((size_t)B * 256 * kNs[1]);
  fpOut[0] = alloc((size_t)B * 128 * kNs[0]);
  float* headTmp = alloc((size_t)B * 64 * kNs[0]);

  const float* lx[5] = {PC, nxw[1], nxw[2], nxw[3], nxw[4]};

  auto launch_mlp = [&](const float* X, const float* Wp, const float* b0,
                        const float* sc, const float* b1, float* Y, int M,
                        int K, int NCOL, int relu) {
    int tiles = ((M + 15) / 16) * (NCOL / 64);
    dim3 g((unsigned)((tiles + 7) / 8), (unsigned)B);
    mlp_wmma_kernel<<<g, 256, 0, stream>>>(X, Wp, b0, sc, b1, Y, M, K, NCOL, relu);
  };

  // ---- Set Abstraction levels ----
  for (int l = 0; l < 4; ++l) {
    fps_kernel<<<B, FPS_T, 0, stream>>>(lx[l], nxw[l + 1], kNs[l], kNs[l + 1]);
    int coff = 0;
    for (int s = 0; s < 2; ++s) {
      const int ns = kNSample[s];
      const int Cin = 3 + kCf[l];
      const int np = kNs[l + 1];
      const float r = kRadius[l][s];
      ballquery_group_kernel<<<dim3((unsigned)np, (unsigned)B), 32, 0, stream>>>(
          lx[l], kNs[l], lx[l + 1], np, lf[l], kCf[l], bufA, ns, r * r, Cin);
      const int NCOL = np * ns;
      const float* Xc = bufA;
      float* bufs[2] = {bufB, bufA};
      for (int li = 0; li < 3; ++li) {
        int M = SA_CH[l][s][li + 1];
        int K = SA_CH[l][s][li];
        float* Yc = bufs[li & 1];
        launch_mlp(Xc, saW[l][s][li], nullptr, saG[l][s][li], saB[l][s][li],
                   Yc, M, K, NCOL, 1);
        Xc = Yc;
      }
      const int Mlast = SA_CH[l][s][3];  // output of layer 3 is in bufB
      maxpool_kernel<<<dim3((unsigned)((Mlast * np + 255) / 256), (unsigned)B),
                       256, 0, stream>>>(bufB, lf[l + 1], Mlast, np, ns,
                                         kCf[l + 1], coff);
      coff += Mlast;
    }
  }

  // ---- Feature Propagation levels (i = -1..-4 -> d = 3..0) ----
  const float* fknown = lf[4];
  int Ck = kCf[4];  // 1024
  for (int d = 3; d >= 0; --d) {
    const int Nu = kNs[d];
    const int Nk = kNs[d + 1];
    const int Cs = (d == 0) ? 0 : kCf[d];
    const int Ccat = FP_CH[d][0];
    if (Cs > 0) {
      copychan_kernel<<<dim3((unsigned)((Cs * Nu + 255) / 256), (unsigned)B),
                        256, 0, stream>>>(lf[d], bufA, Cs, Nu, Ccat);
    }
    three_interp_kernel<<<dim3((unsigned)Nu, (unsigned)B), 32, 0, stream>>>(
        lx[d], Nu, lx[d + 1], Nk, fknown, Ck, bufA, Ccat, Cs);
    launch_mlp(bufA, fpW[d][0], nullptr, fpG[d][0], fpB[d][0], bufB,
               FP_CH[d][1], Ccat, Nu, 1);
    launch_mlp(bufB, fpW[d][1], nullptr, fpG[d][1], fpB[d][1], fpOut[d],
               FP_CH[d][2], FP_CH[d][1], Nu, 1);
    fknown = fpOut[d];
    Ck = FP_CH[d][2];
  }

  // ---- head (no ReLU) ----
  launch_mlp(fpOut[0], hW1, hb1, hg1, hbe1, headTmp, 64, 128, kNs[0], 0);
  launch_mlp(headTmp, hW2, hb2, nullptr, nullptr, bufB, 3, 64, kNs[0], 0);
  transpose_out_kernel<<<dim3((unsigned)(kNs[0] / 256), (unsigned)B), 256, 0,
                         stream>>>(bufB, (float*)d_out, kNs[0]);
}